// Embedding_78228534329818
// MI455X (gfx1250) — compile-verified
//
#include <hip/hip_runtime.h>
#include <stdint.h>

// Embedding gather: out[row, :] = w[x[row], :], row in [0, B*S), D = 1024 fp32.
// One-hot matmul form would be ~537 GFLOP; the gather form is ~64 MB of traffic
// -> HBM-roofline ~2.8 us @ 23.3 TB/s. Implemented with the CDNA5 async
// global<->LDS data-mover path (ASYNCcnt): each wave keeps 8 x 512B bulk
// transfers in flight with zero VGPR data traffic.

#define LDS_AS __attribute__((address_space(3)))
#define GLB_AS __attribute__((address_space(1)))

typedef int   v4i __attribute__((vector_size(16)));  // native int4 (builtin ABI type)
typedef float v4f __attribute__((vector_size(16)));  // native float4

#if __has_builtin(__builtin_amdgcn_global_load_async_to_lds_b128) && \
    __has_builtin(__builtin_amdgcn_global_store_async_from_lds_b128) && \
    __has_builtin(__builtin_amdgcn_s_wait_asynccnt)
#define USE_ASYNC_LDS 1
#else
#define USE_ASYNC_LDS 0
#endif

constexpr int ROWS_PER_BLOCK = 8;   // 8 rows x 4KB = 32KB LDS staging per block
constexpr int D = 1024;             // embedding dim (fp32)

#if USE_ASYNC_LDS

__device__ __forceinline__ GLB_AS v4i* glb_v4(const void* p) {
  return (GLB_AS v4i*)(uintptr_t)p;
}
__device__ __forceinline__ LDS_AS v4i* lds_v4(const void* p) {
  // Generic LDS pointer low 32 bits == LDS byte address (aperture truncation).
  return (LDS_AS v4i*)(uint32_t)(uintptr_t)p;
}

__global__ void __launch_bounds__(256)
embed_gather_async(const int* __restrict__ x, const float* __restrict__ w,
                   float* __restrict__ out, int rows) {
  __shared__ float smem[ROWS_PER_BLOCK][D];

  const int tid  = threadIdx.x;
  const int col  = (tid >> 5) * 128 + (tid & 31) * 4;  // wave-private 512B slice
  const int row0 = blockIdx.x * ROWS_PER_BLOCK;

  if (row0 + ROWS_PER_BLOCK <= rows) {
    // Fast path (all blocks when rows % ROWS_PER_BLOCK == 0): no per-row guards,
    // async ops issue back-to-back on the scalar pipe.
#pragma unroll
    for (int r = 0; r < ROWS_PER_BLOCK; ++r) {
      const int tok = x[row0 + r];               // block-uniform -> s_load
      __builtin_amdgcn_global_load_async_to_lds_b128(
          glb_v4(w + (size_t)tok * D + col), lds_v4(&smem[r][col]), 0, 0);
    }
    // Loads and stores share ASYNCcnt and complete unordered w.r.t. each
    // other, so drain loads before stores read the same LDS. Wave-private
    // slices -> no workgroup barrier needed.
    __builtin_amdgcn_s_wait_asynccnt(0);
#pragma unroll
    for (int r = 0; r < ROWS_PER_BLOCK; ++r) {
      __builtin_amdgcn_global_store_async_from_lds_b128(
          glb_v4(out + (size_t)(row0 + r) * D + col), lds_v4(&smem[r][col]), 0, 0);
    }
    __builtin_amdgcn_s_wait_asynccnt(0);  // S_ENDPGM also waits-idle
  } else {
    // Tail path: per-row uniform guard.
    for (int r = 0; r < ROWS_PER_BLOCK; ++r) {
      const int row = row0 + r;
      if (row < rows) {
        const int tok = x[row];
        __builtin_amdgcn_global_load_async_to_lds_b128(
            glb_v4(w + (size_t)tok * D + col), lds_v4(&smem[r][col]), 0, 0);
      }
    }
    __builtin_amdgcn_s_wait_asynccnt(0);
    for (int r = 0; r < ROWS_PER_BLOCK; ++r) {
      const int row = row0 + r;
      if (row < rows) {
        __builtin_amdgcn_global_store_async_from_lds_b128(
            glb_v4(out + (size_t)row * D + col), lds_v4(&smem[r][col]), 0, 0);
      }
    }
    __builtin_amdgcn_s_wait_asynccnt(0);
  }
}

#else  // !USE_ASYNC_LDS

// Fallback: direct 128-bit gather copy, nontemporal stores (write-once output).
__global__ void __launch_bounds__(256)
embed_gather_direct(const int* __restrict__ x, const float* __restrict__ w,
                    float* __restrict__ out, int rows) {
  const int col  = threadIdx.x * 4;          // 256 threads x 4 floats = 1024
  const int row0 = blockIdx.x * ROWS_PER_BLOCK;
#pragma unroll
  for (int r = 0; r < ROWS_PER_BLOCK; ++r) {
    const int row = row0 + r;
    if (row < rows) {
      const int tok = x[row];
      const v4f v = *(const v4f*)(w + (size_t)tok * D + col);
      __builtin_nontemporal_store(v, (v4f*)(out + (size_t)row * D + col));
    }
  }
}

#endif  // USE_ASYNC_LDS

extern "C" void kernel_launch(void* const* d_in, const int* in_sizes, int n_in,
                              void* d_out, int out_size, void* d_ws, size_t ws_size,
                              hipStream_t stream) {
  const int*   x   = (const int*)d_in[0];    // (4, 2048) token ids
  const float* w   = (const float*)d_in[1];  // (32000, 1024) fp32 table
  float*       out = (float*)d_out;          // (4, 2048, 1024) fp32

  const int rows = in_sizes[0];              // 8192
  const int nblk = (rows + ROWS_PER_BLOCK - 1) / ROWS_PER_BLOCK;

#if USE_ASYNC_LDS
  embed_gather_async<<<nblk, 256, 0, stream>>>(x, w, out, rows);
#else
  embed_gather_direct<<<nblk, 256, 0, stream>>>(x, w, out, rows);
#endif
}